// TrustRegionInverseWUncertainty_20744692040055
// MI455X (gfx1250) — compile-verified
//
#include <hip/hip_runtime.h>
#include <math.h>

#define Bn 8
#define Cn 16
#define Hn 240
#define Wn 320
#define Nn (Hn*Wn)            // 76800
#define NBLK 30               // blocks per batch
#define PIX_PER_BLOCK (Nn/NBLK)     // 2560
#define WAVES 8
#define PIX_PER_WAVE (PIX_PER_BLOCK/WAVES) // 320
#define GROUPS (PIX_PER_WAVE/32)           // 10

typedef __attribute__((ext_vector_type(2))) float v2f;
typedef __attribute__((ext_vector_type(8))) float v8f;

// ---------------- small math helpers ----------------
__device__ __forceinline__ void twist2mat_dev(float w0, float w1, float w2, float* R) {
  float th = sqrtf(w0*w0 + w1*w1 + w2*w2) + 1e-12f;
  float a0 = w0/th, a1 = w1/th, a2 = w2/th;
  float s  = sinf(th);
  float c1 = 1.0f - cosf(th);
  float K[9] = { 0.f, -a2,  a1,
                 a2,  0.f, -a0,
                -a1,  a0,  0.f };
  float K2[9];
  #pragma unroll
  for (int r = 0; r < 3; ++r)
    #pragma unroll
    for (int c = 0; c < 3; ++c)
      K2[r*3+c] = K[r*3+0]*K[0*3+c] + K[r*3+1]*K[1*3+c] + K[r*3+2]*K[2*3+c];
  #pragma unroll
  for (int i = 0; i < 9; ++i) R[i] = s*K[i] + c1*K2[i];
  R[0] += 1.f; R[4] += 1.f; R[8] += 1.f;
}

// ---------------- init: R = twist2mat(twist0[:,:3]), t = twist0[:,3:] ----------------
__global__ void tri_init_kernel(const float* __restrict__ twist0, float* __restrict__ Rt) {
  int b = threadIdx.x;
  if (b >= Bn) return;
  float R[9];
  twist2mat_dev(twist0[b*6+0], twist0[b*6+1], twist0[b*6+2], R);
  #pragma unroll
  for (int i = 0; i < 9; ++i) Rt[b*9 + i] = R[i];
  Rt[Bn*9 + b*3 + 0] = twist0[b*6+3];
  Rt[Bn*9 + b*3 + 1] = twist0[b*6+4];
  Rt[Bn*9 + b*3 + 2] = twist0[b*6+5];
}

// ---------------- zero the per-batch 7x7 Gram accumulators ----------------
__global__ void tri_zero_kernel(float* __restrict__ gram) {
  int i = threadIdx.x;
  if (i < Bn*49) gram[i] = 0.f;
}

// ---------------- main: build rows [Jw(6), r] and WMMA-accumulate G = A^T A ----------------
__global__ __launch_bounds__(256) void tri_gram_kernel(
    const float* __restrict__ x0,     const float* __restrict__ x1,
    const float* __restrict__ invD0,  const float* __restrict__ invD1,
    const float* __restrict__ sigma0, const float* __restrict__ sigma1,
    const float* __restrict__ Kmat,   const float* __restrict__ Rt,
    float* __restrict__ gram)
{
  __shared__ float lds[WAVES][32][17];   // per-wave staging tile, padded stride
  const int tid  = threadIdx.x;
  const int wave = tid >> 5;
  const int lane = tid & 31;
  const int b      = blockIdx.x / NBLK;
  const int blkInB = blockIdx.x % NBLK;

  // pre-zero LDS once: cols 7..15 stay zero forever (WMMA pad columns)
  for (int i = tid; i < WAVES*32*17; i += 256) ((float*)lds)[i] = 0.f;
  __syncthreads();

  const float fx = Kmat[b*4+0], fy = Kmat[b*4+1], cx = Kmat[b*4+2], cy = Kmat[b*4+3];
  const float* Rm = Rt + b*9;
  const float* tm = Rt + Bn*9 + b*3;
  const float R00=Rm[0], R01=Rm[1], R02=Rm[2];
  const float R10=Rm[3], R11=Rm[4], R12=Rm[5];
  const float R20=Rm[6], R21=Rm[7], R22=Rm[8];
  const float t0=tm[0], t1=tm[1], t2=tm[2];

  v8f acc = {0.f,0.f,0.f,0.f,0.f,0.f,0.f,0.f};
  float (*myLds)[17] = lds[wave];
  const int m = lane & 15;
  const int h = lane >> 4;

  for (int g = 0; g < GROUPS; ++g) {
    const int n  = blkInB*PIX_PER_BLOCK + wave*PIX_PER_WAVE + g*32 + lane;
    const int ui = n % Wn, vi = n / Wn;

    // ---- per-pixel warp ----
    const float px = ((float)ui - cx) / fx;
    const float py = ((float)vi - cy) / fy;
    const float d  = invD0[b*Nn + n];
    const float X  = R00*px + R01*py + R02 + t0*d;
    const float Y  = R10*px + R11*py + R12 + t1*d;
    const float S  = R20*px + R21*py + R22 + t2*d;
    const float u_w = X/S*fx + cx;
    const float v_w = Y/S*fy + cy;
    const float inv_z = d / S;

    // ---- bilinear setup (clipped) ----
    const float uc = fminf(fmaxf(u_w, 0.f), (float)(Wn-1));
    const float vc = fminf(fmaxf(v_w, 0.f), (float)(Hn-1));
    const float xf = floorf(uc), yf = floorf(vc);
    const float wx = uc - xf,    wy = vc - yf;
    const int x0i = (int)xf, y0i = (int)yf;
    const int x1i = min(x0i+1, Wn-1), y1i = min(y0i+1, Hn-1);
    const int i00 = y0i*Wn + x0i, i01 = y0i*Wn + x1i;
    const int i10 = y1i*Wn + x0i, i11 = y1i*Wn + x1i;
    const float w00 = (1.f-wx)*(1.f-wy), w01 = wx*(1.f-wy);
    const float w10 = (1.f-wx)*wy,       w11 = wx*wy;

    // ---- occlusion check ----
    const float* D1 = invD1 + b*Nn;
    const float izw = D1[i00]*w00 + D1[i01]*w01 + D1[i10]*w10 + D1[i11]*w11;
    const bool inlier = inv_z > (izw - 0.1f);
    const bool inview = (u_w > 0.f) && (u_w < (float)Wn) && (v_w > 0.f) && (v_w < (float)Hn);
    const bool occ = !(inlier && inview);

    // ---- warp Jacobian (6 per px/py direction) ----
    const float xy = px*py;
    float Jxp[6], Jyp[6];
    Jxp[0] = fx*(-xy);           Jyp[0] = fy*(-(1.f+py*py));
    Jxp[1] = fx*(1.f+px*px);     Jyp[1] = fy*(xy);
    Jxp[2] = fx*(-py);           Jyp[2] = fy*(px);
    Jxp[3] = fx*d;               Jyp[3] = 0.f;
    Jxp[4] = 0.f;                Jyp[4] = fy*d;
    Jxp[5] = fx*(-d*px);         Jyp[5] = fy*(-d*py);

    // sobel neighbor coords (edge padding)
    const int xm = max(ui-1,0), xp = min(ui+1,Wn-1);
    const int ym = max(vi-1,0), yp = min(vi+1,Hn-1);

    for (int c = 0; c < Cn; ++c) {
      const int base = (b*Cn + c)*Nn;
      const float* X0 = x0 + base;
      // ---- sobel gradient of x0 (normalized) ----
      const float tl = X0[ym*Wn+xm], tcv = X0[ym*Wn+ui], tr = X0[ym*Wn+xp];
      const float ml = X0[vi*Wn+xm],                     mr = X0[vi*Wn+xp];
      const float bl = X0[yp*Wn+xm], bcv = X0[yp*Wn+ui], br = X0[yp*Wn+xp];
      const float dx = (tr - tl) + 2.f*(mr - ml) + (br - bl);
      const float dy = (bl - tl) + 2.f*(bcv - tcv) + (br - tr);
      const float rmag = 1.0f / sqrtf(dx*dx + dy*dy + 1e-8f);
      const float jfx = dx*rmag, jfy = dy*rmag;

      // ---- warped samples + residual ----
      const float* X1 = x1 + base;
      const float* S1 = sigma1 + base;
      const float fr = X1[i00]*w00 + X1[i01]*w01 + X1[i10]*w10 + X1[i11]*w11;
      const float sr = S1[i00]*w00 + S1[i01]*w01 + S1[i10]*w10 + S1[i11]*w11;
      const float x0v = X0[n];
      const float s0v = sigma0[base + n];
      const float sig = sqrtf(sr*sr + s0v*s0v);
      const float scale = occ ? 0.f : (1.0f / sig);

      // ---- stage row [Jw0..Jw5, r] into LDS (cols 7..15 stay 0) ----
      #pragma unroll
      for (int k = 0; k < 6; ++k)
        myLds[lane][k] = (jfx*Jxp[k] + jfy*Jyp[k]) * scale;
      myLds[lane][6] = (fr - x0v) * scale;

      __builtin_amdgcn_wave_barrier();      // same-wave LDS is in-order; just stop reordering
      asm volatile("" ::: "memory");

      // ---- 8 x WMMA f32 16x16x4 : G += chunk^T * chunk (A frag == B frag) ----
      #pragma unroll
      for (int j = 0; j < 8; ++j) {
        v2f a;
        a.x = myLds[j*4 + 2*h    ][m];
        a.y = myLds[j*4 + 2*h + 1][m];
        acc = __builtin_amdgcn_wmma_f32_16x16x4_f32(
                  false, a, false, a, (short)0, acc, false, false);
      }

      __builtin_amdgcn_wave_barrier();      // keep next store after this step's loads
      asm volatile("" ::: "memory");
    }
  }

  // ---- reduce partial Gram into global (rows 0..6 live in lanes 0..15, vgprs 0..6) ----
  if (lane < 7) {
    #pragma unroll
    for (int v = 0; v < 7; ++v)
      atomicAdd(&gram[b*49 + v*7 + lane], acc[v]);
  }
}

// ---------------- per-batch 6x6 LM solve + SE(3) update ----------------
__global__ void tri_solve_kernel(const float* __restrict__ gram,
                                 float* __restrict__ Rt,
                                 float* __restrict__ out)
{
  int b = threadIdx.x;
  if (b >= Bn) return;

  float Hm[6][7];
  for (int i = 0; i < 6; ++i)
    for (int j = 0; j < 7; ++j)
      Hm[i][j] = gram[b*49 + i*7 + j];   // col 6 = Rhs

  float trc = 0.f;
  for (int i = 0; i < 6; ++i) trc += Hm[i][i];
  for (int i = 0; i < 6; ++i) Hm[i][i] += trc * 1e-6f;

  // Gauss-Jordan (H is SPD + damping)
  for (int k = 0; k < 6; ++k) {
    float inv = 1.0f / Hm[k][k];
    for (int j = k; j < 7; ++j) Hm[k][j] *= inv;
    for (int i = 0; i < 6; ++i) {
      if (i == k) continue;
      float f = Hm[i][k];
      for (int j = k; j < 7; ++j) Hm[i][j] -= f * Hm[k][j];
    }
  }
  float xi[6];
  for (int i = 0; i < 6; ++i) xi[i] = Hm[i][6];

  // d_R = twist2mat(-xi[0:3]); d_t = -(d_R @ xi[3:6])
  float dR[9];
  twist2mat_dev(-xi[0], -xi[1], -xi[2], dR);
  float dt0 = -(dR[0]*xi[3] + dR[1]*xi[4] + dR[2]*xi[5]);
  float dt1 = -(dR[3]*xi[3] + dR[4]*xi[4] + dR[5]*xi[5]);
  float dt2 = -(dR[6]*xi[3] + dR[7]*xi[4] + dR[8]*xi[5]);

  float R[9];
  for (int i = 0; i < 9; ++i) R[i] = Rt[b*9 + i];
  float t0 = Rt[Bn*9 + b*3 + 0];
  float t1 = Rt[Bn*9 + b*3 + 1];
  float t2 = Rt[Bn*9 + b*3 + 2];

  // t = R @ d_t + t
  float tn0 = R[0]*dt0 + R[1]*dt1 + R[2]*dt2 + t0;
  float tn1 = R[3]*dt0 + R[4]*dt1 + R[5]*dt2 + t1;
  float tn2 = R[6]*dt0 + R[7]*dt1 + R[8]*dt2 + t2;

  // R = R @ d_R
  float Rn[9];
  for (int r = 0; r < 3; ++r)
    for (int c = 0; c < 3; ++c)
      Rn[r*3+c] = R[r*3+0]*dR[0*3+c] + R[r*3+1]*dR[1*3+c] + R[r*3+2]*dR[2*3+c];

  for (int i = 0; i < 9; ++i) Rt[b*9 + i] = Rn[i];
  Rt[Bn*9 + b*3 + 0] = tn0;
  Rt[Bn*9 + b*3 + 1] = tn1;
  Rt[Bn*9 + b*3 + 2] = tn2;

  // out = concat([R.reshape(B,9), t], axis=1)
  for (int i = 0; i < 9; ++i) out[b*12 + i] = Rn[i];
  out[b*12 +  9] = tn0;
  out[b*12 + 10] = tn1;
  out[b*12 + 11] = tn2;
}

extern "C" void kernel_launch(void* const* d_in, const int* in_sizes, int n_in,
                              void* d_out, int out_size, void* d_ws, size_t ws_size,
                              hipStream_t stream) {
  const float* twist0 = (const float*)d_in[0];
  const float* x0     = (const float*)d_in[1];
  const float* x1     = (const float*)d_in[2];
  const float* invD0  = (const float*)d_in[3];
  const float* invD1  = (const float*)d_in[4];
  const float* sigma0 = (const float*)d_in[5];
  const float* sigma1 = (const float*)d_in[6];
  const float* Kmat   = (const float*)d_in[7];

  float* ws   = (float*)d_ws;
  float* Rt   = ws;            // B*9 (R) + B*3 (t)
  float* gram = ws + Bn*12;    // B*49

  tri_init_kernel<<<1, 32, 0, stream>>>(twist0, Rt);
  for (int it = 0; it < 3; ++it) {
    tri_zero_kernel<<<1, 512, 0, stream>>>(gram);
    tri_gram_kernel<<<Bn*NBLK, 256, 0, stream>>>(x0, x1, invD0, invD1,
                                                 sigma0, sigma1, Kmat, Rt, gram);
    tri_solve_kernel<<<1, 32, 0, stream>>>(gram, Rt, (float*)d_out);
  }
}